// QueryGatedSelfAttention_63127429317124
// MI455X (gfx1250) — compile-verified
//
#include <hip/hip_runtime.h>
#include <hip/hip_bf16.h>

// QueryGatedSelfAttention for MI455X (gfx1250, wave32).
// B=4, N(=NP=NQ)=256, D=512, T=22.627417 (= sqrt(512)).
//
// Cost model: GEMMs ~0.54 GFLOP (negligible); 512MB output write dominates
// (~22us at 23.3 TB/s). So: f32 WMMA for exact-precision score GEMMs, and a
// wide vectorized non-temporal streaming kernel for the gated output.

#define BATCH 4
#define NN    256
#define DD    512
#define INV_T 0.044194173824159216f   // 1/22.627417

typedef __attribute__((ext_vector_type(2))) float v2f;
typedef __attribute__((ext_vector_type(4))) float v4f;
typedef __attribute__((ext_vector_type(8))) float v8f;

// One block = 256 threads = 8 waves, handles a 16-row tile of scores for one
// batch. Each wave computes two 16x16 score tiles via v_wmma_f32_16x16x4_f32
// (K-loop over D=512), stages them in LDS, then does the row softmax with
// wave32 shfl reductions (each wave owns 2 rows of 256).
template<bool WRITE_LOG>
__global__ void qgsa_scores_softmax(const float* __restrict__ Qm,   // (B,N,D) rows
                                    const float* __restrict__ Km,   // (B,N,D) cols
                                    float* __restrict__ attn,       // (B,N,N)
                                    float* __restrict__ log_attn)   // (B,N,N) or unused
{
    __shared__ float smem[16][NN];          // 16 KB

    const int b    = blockIdx.x >> 4;       // 16 row-tiles per batch
    const int rt   = blockIdx.x & 15;
    const int tid  = threadIdx.x;
    const int wave = tid >> 5;
    const int lane = tid & 31;
    const int half = lane >> 4;             // K-half selector (ISA A/B layout)
    const int lm   = lane & 15;             // M (for A) / N (for B) within tile

    const v2f* q2 = (const v2f*)(Qm + ((size_t)b * NN + rt * 16 + lm) * DD);

    #pragma unroll
    for (int t = 0; t < 2; ++t) {
        const int ct = wave * 2 + t;        // column tile 0..15
        const v2f* k2 = (const v2f*)(Km + ((size_t)b * NN + ct * 16 + lm) * DD);
        v8f c = {};
        // D = 512 -> 128 WMMA steps of K=4. Lane loads are contiguous b64.
        for (int kk = 0; kk < DD / 2; kk += 2) {
            v2f a  = q2[kk + half];
            v2f bb = k2[kk + half];
            c = __builtin_amdgcn_wmma_f32_16x16x4_f32(
                    false, a, false, bb, (short)0, c, false, false);
        }
        // C/D layout: element i of lane -> M = i + 8*half, N = lm.
        #pragma unroll
        for (int i = 0; i < 8; ++i)
            smem[i + 8 * half][ct * 16 + lm] = c[i] * INV_T;
    }
    __syncthreads();

    // Softmax: wave w owns rows 2w and 2w+1; lane covers 8 strided columns.
    #pragma unroll
    for (int rr = 0; rr < 2; ++rr) {
        const int r = wave * 2 + rr;
        float vals[8];
        float m = -INFINITY;
        #pragma unroll
        for (int j = 0; j < 8; ++j) {
            vals[j] = smem[r][lane + 32 * j];
            m = fmaxf(m, vals[j]);
        }
        #pragma unroll
        for (int off = 16; off >= 1; off >>= 1)
            m = fmaxf(m, __shfl_xor(m, off, 32));
        float s = 0.0f;
        #pragma unroll
        for (int j = 0; j < 8; ++j) {
            vals[j] = __expf(vals[j] - m);
            s += vals[j];
        }
        #pragma unroll
        for (int off = 16; off >= 1; off >>= 1)
            s += __shfl_xor(s, off, 32);
        const float inv = 1.0f / s;
        const float lse = __logf(s);
        const size_t base = ((size_t)b * NN + rt * 16 + r) * NN;
        #pragma unroll
        for (int j = 0; j < 8; ++j) {
            const int cidx = lane + 32 * j;
            attn[base + cidx] = vals[j] * inv;
            if (WRITE_LOG)
                log_attn[base + cidx] = (smem[r][cidx] - m) - lse;
        }
    }
}

// output[b,n,p,d] = gate[b,n,p] * (v[b,p,d] + query_v[b,n,d])
// Pure streaming kernel: each block covers two (b,n,p) rows of D=512;
// each thread handles one float4. Non-temporal 128-bit stores (512MB >> L2).
__global__ void qgsa_gate_combine(const float* __restrict__ V,
                                  const float* __restrict__ QV,
                                  const float* __restrict__ G,
                                  float* __restrict__ out)
{
    const int tid = threadIdx.x;
    const long long flat = (long long)blockIdx.x * 2 + (tid >> 7); // b*N*N + n*N + p
    const int d = (tid & 127) << 2;                                // d index (x4)

    const int p        = (int)(flat & (NN - 1));
    const long long bn = flat >> 8;                                // b*N + n
    const int bb       = (int)(bn >> 8);

    const float g = G[flat];
    const v4f vv = *(const v4f*)(V  + ((size_t)bb * NN + p) * DD + d);
    const v4f qq = *(const v4f*)(QV + (size_t)bn * DD + d);

    v4f o = g * (vv + qq);
    __builtin_nontemporal_store(o, (v4f*)(out + (size_t)flat * DD + d));
}

extern "C" void kernel_launch(void* const* d_in, const int* in_sizes, int n_in,
                              void* d_out, int out_size, void* d_ws, size_t ws_size,
                              hipStream_t stream) {
    const float* q       = (const float*)d_in[0];
    const float* k       = (const float*)d_in[1];
    const float* v       = (const float*)d_in[2];
    // d_in[3] = 'query' is unused by the reference computation.
    const float* query_q = (const float*)d_in[4];
    const float* query_k = (const float*)d_in[5];
    const float* query_v = (const float*)d_in[6];

    float* out  = (float*)d_out;                                    // (B,N,N,D)
    float* attn = out  + (size_t)BATCH * NN * NN * DD;              // (B,N,N)
    float* logp = attn + (size_t)BATCH * NN * NN;                   // (B,N,N)
    float* gate = (float*)d_ws;                                     // needs 1 MB

    // attn + log_attn (writes both)
    qgsa_scores_softmax<true><<<BATCH * (NN / 16), 256, 0, stream>>>(q, k, attn, logp);
    // query_attn -> workspace (gate), probs only
    qgsa_scores_softmax<false><<<BATCH * (NN / 16), 256, 0, stream>>>(query_q, query_k, gate, gate);
    // 512 MB streaming output
    qgsa_gate_combine<<<(BATCH * NN * NN) / 2, 256, 0, stream>>>(v, query_v, gate, out);
}